// CFModule_12575664243188
// MI455X (gfx1250) — compile-verified
//
#include <hip/hip_runtime.h>
#include <math.h>

typedef float v2f __attribute__((ext_vector_type(2)));
typedef float v8f __attribute__((ext_vector_type(8)));

namespace {
constexpr int kB   = 16;
constexpr int kH   = 256;
constexpr int kW   = 256;
constexpr int kC   = 64;
constexpr int kPos = 16;            // 4x4 pooled positions
constexpr int kNPix = kH * kW;      // 65536 pixels per batch
constexpr int kPixPerBlock = 256;   // 16 waves * 16 pixel-rows
}

// ---------------------------------------------------------------------------
// Kernel 1: AdaptiveAvgPool2d((4,4)) over [B,H,W,C] -> pooled[B][16][C]
// One block per (b, window). 1024 threads: c = tid&63, pixel-group g = tid>>6.
// ---------------------------------------------------------------------------
__global__ __launch_bounds__(1024)
void pool4x4_kernel(const float* __restrict__ x, float* __restrict__ pooled) {
  __shared__ float part[16 * kC];
  const int w   = blockIdx.x;       // 0..255
  const int b   = w >> 4;
  const int pos = w & 15;           // ph*4 + pw
  const int ph  = pos >> 2, pw = pos & 3;
  const int tid = threadIdx.x;
  const int c   = tid & 63;
  const int g   = tid >> 6;         // 0..15, each group covers 256 pixels

  const size_t rowbase = ((size_t)b * kH + (size_t)ph * 64) * kW + (size_t)pw * 64;
  float s = 0.f;
  #pragma unroll 4
  for (int qq = 0; qq < 256; ++qq) {
    const int q   = g * 256 + qq;   // window-local pixel 0..4095
    const int row = q >> 6;
    const int col = q & 63;
    s += x[(rowbase + (size_t)row * kW + col) * kC + c];
  }
  part[g * kC + c] = s;
  __syncthreads();
  if (tid < kC) {
    float t = 0.f;
    #pragma unroll
    for (int gg = 0; gg < 16; ++gg) t += part[gg * kC + tid];
    pooled[(b * kPos + pos) * kC + tid] = t * (1.0f / 4096.0f);
  }
}

// ---------------------------------------------------------------------------
// Kernel 2: dots = 0.25 * xf xf^T, softmax rows, store TRANSPOSED:
//   attnT[b][c][t] = softmax_c(dots)[t][c]
// One block per batch, 64 threads (thread i owns output row t=i).
// ---------------------------------------------------------------------------
__global__ __launch_bounds__(64)
void attn_kernel(const float* __restrict__ pooled, float* __restrict__ attnT) {
  __shared__ float pf[kPos * kC];   // [n][c]
  __shared__ float dm[kC * kC];     // row i = dots[i][*]
  const int b = blockIdx.x;
  const int i = threadIdx.x;        // 0..63

  for (int n = 0; n < kPos; ++n)
    pf[n * kC + i] = pooled[(b * kPos + n) * kC + i];
  __syncthreads();

  for (int j = 0; j < kC; ++j) {
    float s = 0.f;
    #pragma unroll
    for (int n = 0; n < kPos; ++n) s += pf[n * kC + i] * pf[n * kC + j];
    dm[i * kC + j] = s * 0.25f;     // scale = 16^-0.5
  }
  // softmax over row i (thread-private row, no barrier needed)
  float m = -INFINITY;
  for (int j = 0; j < kC; ++j) m = fmaxf(m, dm[i * kC + j]);
  float ssum = 0.f;
  for (int j = 0; j < kC; ++j) {
    const float e = expf(dm[i * kC + j] - m);
    dm[i * kC + j] = e;
    ssum += e;
  }
  const float inv = 1.0f / ssum;
  for (int j = 0; j < kC; ++j)
    attnT[((size_t)b * kC + j) * kC + i] = dm[i * kC + j] * inv;  // [c=j][t=i]
}

// ---------------------------------------------------------------------------
// Kernel 3: out[b,p,t] = GELU( sum_c x[b,p,c] * attn[b,t,c] )
// GEMM per batch: X[65536 x 64] * attnT[64 x 64] via v_wmma_f32_16x16x4_f32.
// 512 threads = 16 waves; each wave produces a 16-pixel x 64-channel tile.
//   A frag (16x4 f32): lane: M=lane&15, K0=2*(lane>>4) -> v2f from global x
//   B frag (4x16 f32): lane: N=lane&15, K0=2*(lane>>4) -> from LDS attnT
//   D (16x16 f32, 4 N-tiles): vgpr r: M = r + 8*(lane>>4), N = lane&15
// ---------------------------------------------------------------------------
__global__ __launch_bounds__(512)
void apply_attn_gelu_kernel(const float* __restrict__ x,
                            const float* __restrict__ attnT,
                            float* __restrict__ out) {
  __shared__ float at[kC * kC];     // attnT for this batch: [c][t], 16 KB
  const int blk  = blockIdx.x;
  const int b    = blk >> 8;        // 256 blocks per batch
  const int pb   = blk & 255;
  const int tid  = threadIdx.x;
  const int wave = tid >> 5;
  const int lane = tid & 31;
  const int M    = lane & 15;       // also N for B/D frags
  const int half = lane >> 4;       // 0/1
  const int K0   = half * 2;
  const int p0   = pb * kPixPerBlock + wave * 16;

  const float* xrow = x + ((size_t)b * kNPix + p0 + M) * kC;  // this lane's pixel row
  __builtin_prefetch(xrow, 0, 0);   // global_prefetch_b8: warm x tile during LDS fill

  // stage this batch's attnT into LDS (coalesced, 8 iters exactly -> no divergence)
  for (int k = tid; k < kC * kC; k += 512) at[k] = attnT[(size_t)b * kC * kC + k];
  __syncthreads();

  v8f acc[4] = {v8f{}, v8f{}, v8f{}, v8f{}};

  #pragma unroll
  for (int ks = 0; ks < 16; ++ks) {           // K = 64 in steps of 4
    const int kb = ks * 4;
    const v2f a = *(const v2f*)(xrow + kb + K0);          // A: X[M][kb+K0 .. +1]
    #pragma unroll
    for (int nt = 0; nt < 4; ++nt) {                      // N = 64 in 16-wide tiles
      v2f bb;
      bb.x = at[(kb + K0)     * kC + nt * 16 + M];        // B[k][n] = attnT[k][n0+n]
      bb.y = at[(kb + K0 + 1) * kC + nt * 16 + M];
      acc[nt] = __builtin_amdgcn_wmma_f32_16x16x4_f32(
          /*neg_a=*/false, a, /*neg_b=*/false, bb,
          /*c_mod=*/(short)0, acc[nt], /*reuse_a=*/false, /*reuse_b=*/false);
    }
  }

  // epilogue: exact-erf GELU, scatter per D layout
  float* obase = out + (size_t)b * kNPix * kC;
  #pragma unroll
  for (int r = 0; r < 8; ++r) {
    const int prow = p0 + r + half * 8;
    const size_t o = (size_t)prow * kC + M;
    #pragma unroll
    for (int nt = 0; nt < 4; ++nt) {
      const float v = acc[nt][r];
      obase[o + nt * 16] = 0.5f * v * (1.0f + erff(v * 0.70710678118654752f));
    }
  }
}

// ---------------------------------------------------------------------------
extern "C" void kernel_launch(void* const* d_in, const int* in_sizes, int n_in,
                              void* d_out, int out_size, void* d_ws, size_t ws_size,
                              hipStream_t stream) {
  (void)in_sizes; (void)n_in; (void)out_size; (void)ws_size;
  const float* x = (const float*)d_in[0];
  float* out    = (float*)d_out;
  float* pooled = (float*)d_ws;                 // 16*16*64 floats   = 64 KB
  float* attnT  = pooled + kB * kPos * kC;      // 16*64*64 floats   = 256 KB

  pool4x4_kernel<<<kB * kPos, 1024, 0, stream>>>(x, pooled);
  attn_kernel<<<kB, 64, 0, stream>>>(pooled, attnT);
  apply_attn_gelu_kernel<<<kB * (kNPix / kPixPerBlock), 512, 0, stream>>>(x, attnT, out);
}